// ContrastLoss_4269197492347
// MI455X (gfx1250) — compile-verified
//
#include <hip/hip_runtime.h>
#include <hip/hip_bf16.h>
#include <math.h>

typedef __attribute__((ext_vector_type(2))) float v2f;
typedef __attribute__((ext_vector_type(8))) float v8f;

#define BDIM 16
#define KCLS 5
#define CDIM 256
#define HWN  16384
#define QDIM 64

// ---------------- Kernel Z: zero the atomic accumulators (counts) ----------
__global__ void cl_zero_kernel(float* counts) {
    int t = threadIdx.x;
    if (t < BDIM * KCLS) counts[t] = 0.0f;
}

// ---------------- Kernel A: pred (argmax / gt) + class counts --------------
__global__ __launch_bounds__(256) void cl_pred_kernel(
    const float* __restrict__ res1, const int* __restrict__ gt,
    const int* __restrict__ label_bs, unsigned char* __restrict__ pred,
    float* __restrict__ counts) {
    __shared__ int hist[KCLS];
    const int blocksPerB = HWN / 256;
    int b = blockIdx.x / blocksPerB;
    int p = (blockIdx.x % blocksPerB) * 256 + threadIdx.x;
    if (threadIdx.x < KCLS) hist[threadIdx.x] = 0;
    __syncthreads();

    int pr;
    if (b < *label_bs) {
        pr = gt[b * HWN + p];
    } else {
        const float* r = res1 + (size_t)b * KCLS * HWN + p;
        float best = r[0];
        pr = 0;
        #pragma unroll
        for (int k = 1; k < KCLS; ++k) {
            float v = r[(size_t)k * HWN];
            if (v > best) { best = v; pr = k; }   // first-max wins (strict >)
        }
    }
    pred[b * HWN + p] = (unsigned char)pr;
    atomicAdd(&hist[pr], 1);
    __syncthreads();
    if (threadIdx.x < KCLS)
        atomicAdd(&counts[b * KCLS + threadIdx.x], (float)hist[threadIdx.x]);
}

// ---------------- Kernel B: per-class channel sums (bandwidth kernel) ------
// Block = (b, group of 16 channels). pred[b,:] cached in LDS once; fea
// streamed with non-temporal loads (streams 268MB > L2, never reused).
#define CG 16
__global__ __launch_bounds__(256) void cl_segsum_kernel(
    const float* __restrict__ fea, const unsigned char* __restrict__ pred,
    float* __restrict__ sums) {
    __shared__ unsigned char predl[HWN];
    __shared__ float red[8 * KCLS];
    int b  = blockIdx.x / (CDIM / CG);
    int cg = blockIdx.x % (CDIM / CG);
    int tid = threadIdx.x;

    // cooperative 16B-wide copy of pred[b,:] into LDS
    const uint4* ps = (const uint4*)(pred + (size_t)b * HWN);
    uint4* pd = (uint4*)predl;
    for (int i = tid; i < HWN / 16; i += 256) pd[i] = ps[i];
    __syncthreads();

    for (int cc = 0; cc < CG; ++cc) {
        int c = cg * CG + cc;
        const float* f = fea + ((size_t)b * CDIM + c) * HWN;
        float a0 = 0.f, a1 = 0.f, a2 = 0.f, a3 = 0.f, a4 = 0.f;
        for (int p = tid; p < HWN; p += 256) {
            float v = __builtin_nontemporal_load(f + p);
            int pr = predl[p];
            a0 += (pr == 0) ? v : 0.f;
            a1 += (pr == 1) ? v : 0.f;
            a2 += (pr == 2) ? v : 0.f;
            a3 += (pr == 3) ? v : 0.f;
            a4 += (pr == 4) ? v : 0.f;
        }
        // wave32 reduction
        #pragma unroll
        for (int off = 16; off > 0; off >>= 1) {
            a0 += __shfl_down(a0, off, 32);
            a1 += __shfl_down(a1, off, 32);
            a2 += __shfl_down(a2, off, 32);
            a3 += __shfl_down(a3, off, 32);
            a4 += __shfl_down(a4, off, 32);
        }
        int wave = tid >> 5;
        if ((tid & 31) == 0) {
            red[wave * KCLS + 0] = a0;
            red[wave * KCLS + 1] = a1;
            red[wave * KCLS + 2] = a2;
            red[wave * KCLS + 3] = a3;
            red[wave * KCLS + 4] = a4;
        }
        __syncthreads();
        if (tid < KCLS) {
            float s = 0.f;
            #pragma unroll
            for (int w = 0; w < 8; ++w) s += red[w * KCLS + tid];
            sums[((size_t)b * KCLS + tid) * CDIM + c] = s;
        }
        __syncthreads();
    }
}

// ---------------- Kernel C: normalize, prev-present-class select, query ----
__global__ __launch_bounds__(256) void cl_finalize_kernel(
    const float* __restrict__ counts, const float* __restrict__ sums,
    float* __restrict__ keys, float* __restrict__ query,
    float* __restrict__ validf, float* __restrict__ wb) {
    __shared__ int   prevsel[BDIM][KCLS];
    __shared__ float validl[BDIM][KCLS];
    int tid = threadIdx.x;

    if (tid < BDIM * KCLS) {   // one thread per (b,k): mean + L2 norm
        float cnt = counts[tid];
        float inv = 1.0f / fmaxf(cnt, 1.0f);
        const float* s = sums + (size_t)tid * CDIM;
        float nrm = 0.0f;
        for (int c = 0; c < CDIM; ++c) { float m = s[c] * inv; nrm += m * m; }
        nrm = sqrtf(nrm);
        float scale = inv / fmaxf(nrm, 1e-12f);
        float* kk = keys + (size_t)tid * CDIM;
        for (int c = 0; c < CDIM; ++c) kk[c] = s[c] * scale;
    }
    if (tid < BDIM) {          // one thread per b: prev / valid / denom
        int b = tid;
        int cm = -1, smallest = -1, np = 0;
        int  prevA[KCLS];
        bool pres[KCLS];
        #pragma unroll
        for (int k = 0; k < KCLS; ++k) {
            pres[k]  = counts[b * KCLS + k] > 0.0f;
            prevA[k] = cm;                      // cummax(idx)[k-1]
            if (pres[k]) { if (smallest < 0) smallest = k; np++; cm = k; }
        }
        #pragma unroll
        for (int k = 0; k < KCLS; ++k) {
            bool valid = pres[k] && (k != smallest) && (prevA[k] >= 0);
            prevsel[b][k] = (prevA[k] < 0) ? 0 : prevA[k];
            validl[b][k]  = valid ? 1.0f : 0.0f;
        }
        int denom = np - 1;
        wb[b] = (denom > 0) ? 1.0f / (float)denom : 0.0f;
    }
    __syncthreads();

    const float invT = 5.0f;   // 1 / 0.2 ; fold temperature into query
    for (int i = tid; i < BDIM * KCLS; i += 256)
        validf[i] = validl[i / KCLS][i % KCLS];
    for (int i = tid; i < BDIM * KCLS * CDIM; i += 256) {
        int r = i >> 8, c = i & (CDIM - 1);
        int b = r / KCLS, k = r % KCLS;
        query[i] = keys[((size_t)b * KCLS + prevsel[b][k]) * CDIM + c] * invT;
    }
}

// ---------------- Kernel D: sim = query x queues via FP32 WMMA -------------
// rows = B*K = 80 = 5 x 16-tiles, cols = K*Q = 320 = 20 x 16-tiles,
// inner C = 256 = 64 steps of V_WMMA_F32_16X16X4_F32. One wave per tile.
__global__ __launch_bounds__(32) void cl_sim_wmma_kernel(
    const float* __restrict__ query, const float* __restrict__ queues,
    float* __restrict__ sim) {
    int tile = blockIdx.x;          // 0..99
    int rt = tile / 20;             // row tile
    int ct = tile % 20;             // col tile
    int lane = threadIdx.x;
    int r0  = rt * 16;
    int cc0 = ct * 16;
    int m    = lane & 15;           // A row in tile
    int koff = (lane >> 4) * 2;     // A/B inner-K pair held by this lane half
    int n    = lane & 15;           // B/D col in tile
    int j    = cc0 / QDIM;          // 16-col tile never straddles a j block
    int q    = (cc0 % QDIM) + n;

    const float* arow = query + (size_t)(r0 + m) * CDIM;
    v8f acc = {};
    for (int c0 = 0; c0 < CDIM; c0 += 4) {
        v2f a, bfr;
        a.x   = arow[c0 + koff];
        a.y   = arow[c0 + koff + 1];
        bfr.x = queues[((size_t)j * CDIM + c0 + koff) * QDIM + q];
        bfr.y = queues[((size_t)j * CDIM + c0 + koff + 1) * QDIM + q];
        acc = __builtin_amdgcn_wmma_f32_16x16x4_f32(
            /*neg_a=*/false, a, /*neg_b=*/false, bfr,
            /*c_mod=*/(short)0, acc, /*reuse_a=*/false, /*reuse_b=*/false);
    }
    // D layout: VGPR v -> row = r0 + v + (lane>=16 ? 8 : 0), col = lane&15
    int rbase = r0 + ((lane >> 4) * 8);
    #pragma unroll
    for (int v = 0; v < 8; ++v)
        sim[(size_t)(rbase + v) * (KCLS * QDIM) + cc0 + n] = acc[v];
}

// ---------------- Kernel E: exp/log reduction to scalar loss ---------------
__global__ __launch_bounds__(128) void cl_loss_kernel(
    const float* __restrict__ sim, const float* __restrict__ validf,
    const float* __restrict__ wb, float* __restrict__ out) {
    __shared__ float red[128];
    int tid = threadIdx.x;
    float contrib = 0.0f;
    if (tid < BDIM * KCLS) {
        int b = tid / KCLS, k = tid % KCLS;
        const float* row = sim + (size_t)tid * (KCLS * QDIM);
        float neg = 0.0f;
        for (int j = 0; j < KCLS; ++j) {
            if (j == k) continue;
            float s = 0.0f;
            for (int q = 0; q < QDIM; ++q) s += expf(row[j * QDIM + q]);
            neg += s;
        }
        float lsum = 0.0f;
        for (int q = 0; q < QDIM; ++q) {
            float lp = row[k * QDIM + q];
            lsum += lp - logf(expf(lp) + neg);
        }
        float log_prob = lsum * (1.0f / (float)QDIM);
        contrib = -log_prob * validf[tid] * wb[b];
    }
    red[tid] = contrib;
    __syncthreads();
    for (int s = 64; s > 0; s >>= 1) {
        if (tid < s) red[tid] += red[tid + s];
        __syncthreads();
    }
    if (tid == 0) out[0] = red[0] * (1.0f / (float)BDIM);
}

// ---------------- Launch ---------------------------------------------------
extern "C" void kernel_launch(void* const* d_in, const int* in_sizes, int n_in,
                              void* d_out, int out_size, void* d_ws, size_t ws_size,
                              hipStream_t stream) {
    const float* res1    = (const float*)d_in[0];
    const float* fea1    = (const float*)d_in[1];
    const float* queues  = (const float*)d_in[2];
    const int*   gt      = (const int*)d_in[3];
    const int*   label_b = (const int*)d_in[4];
    float* out = (float*)d_out;

    // workspace layout
    unsigned char* pred = (unsigned char*)d_ws;                  // 256 KB
    float* wf     = (float*)((char*)d_ws + (size_t)BDIM * HWN);  // float scratch
    float* counts = wf;                        // 80   (padded to 128)
    float* sums   = wf + 128;                  // 20480
    float* keys   = sums + BDIM * KCLS * CDIM; // 20480
    float* query  = keys + BDIM * KCLS * CDIM; // 20480
    float* sim    = query + BDIM * KCLS * CDIM;// 25600
    float* validf = sim + BDIM * KCLS * KCLS * QDIM; // 80
    float* wbv    = validf + BDIM * KCLS;            // 16

    cl_zero_kernel<<<1, 128, 0, stream>>>(counts);
    cl_pred_kernel<<<BDIM * (HWN / 256), 256, 0, stream>>>(
        res1, gt, label_b, pred, counts);
    cl_segsum_kernel<<<BDIM * (CDIM / CG), 256, 0, stream>>>(fea1, pred, sums);
    cl_finalize_kernel<<<1, 256, 0, stream>>>(counts, sums, keys, query,
                                              validf, wbv);
    cl_sim_wmma_kernel<<<5 * 20, 32, 0, stream>>>(query, queues, sim);
    cl_loss_kernel<<<1, 128, 0, stream>>>(sim, validf, wbv, out);
}